// Self_Attention_Transfer_InstanceSeg_Loss_53171695125041
// MI455X (gfx1250) — compile-verified
//
#include <hip/hip_runtime.h>
#include <hip/hip_bf16.h>

typedef __attribute__((ext_vector_type(2))) float v2f;
typedef __attribute__((ext_vector_type(8))) float v8f;

// ---------------- problem constants ----------------
// scales: 128,64,32,16 -> L = 16384,4096,1024,256 ; Heads = 2,3,5,8 ; B=2, N=8, K=256
// workspace layout (in floats):
//   maskF  @ 0      : per-scale [b][16 rows][L] f32 0/1 masks, rows 8..15 zero-padded
//                     so WMMA A-fragment loads need no lane predication.
//                     bases: 0, 524288, 655360, 688128  (total 696320 floats)
//   S_old  @ 696320 : 576 tiles x 256 f32  (147456 floats)
//   S_new  @ 843776 : 576 tiles x 256 f32  (147456 floats)
// total 991232 floats = ~3.8 MB of ws.
#define MB0 0
#define MB1 524288
#define MB2 655360
#define MB3 688128
#define S_OLD_OFF 696320
#define S_NEW_OFF 843776

// ---------------------------------------------------------------
// Kernel 1: nearest-neighbor resize of bool mask -> f32 0/1 per scale,
// padded to 16 instance rows (rows 8..15 = 0).
// grid = 128 blocks: blockIdx = s*32 + b*16 + r.
// sc divides 512 exactly, so ih = i*(512/sc) == (i*512)//sc.
// ---------------------------------------------------------------
__global__ __launch_bounds__(256)
void sat_mask_prep(const int* __restrict__ mask, float* __restrict__ maskF) {
  const int SC[4] = {128, 64, 32, 16};
  const int LB[4] = {MB0, MB1, MB2, MB3};
  int s  = blockIdx.x >> 5;
  int br = blockIdx.x & 31;            // b*16 + r
  int b  = br >> 4;
  int r  = br & 15;
  int sc = SC[s];
  int L  = sc * sc;
  int step = 512 / sc;
  float* out = maskF + LB[s] + (size_t)br * L;
  if (r < 8) {
    const int* mrow = mask + (size_t)(b * 8 + r) * 512 * 512;
    for (int l = threadIdx.x; l < L; l += blockDim.x) {
      int i  = l / sc;
      int jc = l - i * sc;
      out[l] = (mrow[i * step * 512 + jc * step] != 0) ? 1.0f : 0.0f;
    }
  } else {
    for (int l = threadIdx.x; l < L; l += blockDim.x) out[l] = 0.0f;
  }
}

// ---------------------------------------------------------------
// Kernel 2: masked sums via V_WMMA_F32_16X16X4_F32.
//   D[m=instance, n=channel] += A[m, kk=l] * B[kk=l, n]
// one block per (b, h, ktile); 8 waves split the L contraction (K=8 per
// iteration, unrolled x2), then deterministic LDS reduce across waves.
// A rows 8..15 are zero by construction (padded maskF) -> no predication.
// Per-lane layout (lane = 16*hi + j):
//   A: x = mask[j, l+2hi], y = mask[j, l+2hi+1]
//   B: x = X[l+2hi, chan0+j], y = X[l+2hi+1, chan0+j]
//   C/D: vgpr r holds D[r + 8*hi, j]
// ---------------------------------------------------------------
__global__ __launch_bounds__(256)
void sat_wmma(const float* __restrict__ Xo, const float* __restrict__ Xn,
              const float* __restrict__ MF, float* __restrict__ S_old,
              float* __restrict__ S_new, int H, int L, int tileBase) {
  __shared__ float red[2][8][256];

  int kt = blockIdx.x & 15;          // channel tile (16 tiles of 16 chans)
  int bh = blockIdx.x >> 4;          // b*H + h
  int b  = bh / H;

  int lane = threadIdx.x & 31;
  int wave = threadIdx.x >> 5;
  int j  = lane & 15;
  int hi = lane >> 4;

  const float* mrow = MF + (size_t)(b * 16 + j) * (size_t)L;   // padded rows: j>=8 -> zeros
  size_t xoff = (size_t)bh * (size_t)L * 256 + (size_t)(kt * 16 + j);
  const float* xo = Xo + xoff;
  const float* xn = Xn + xoff;

  v8f co = {};
  v8f cn = {};
  // L is always a multiple of 64; 8 waves x 8 l-values per iteration.
  for (int l0 = wave * 8; l0 < L; l0 += 64) {
    int la = l0 + (hi << 1);
    v2f a0; a0.x = mrow[la];     a0.y = mrow[la + 1];
    v2f a1; a1.x = mrow[la + 4]; a1.y = mrow[la + 5];
    size_t o = (size_t)la * 256;
    v2f bo0; bo0.x = xo[o];        bo0.y = xo[o + 256];
    v2f bo1; bo1.x = xo[o + 1024]; bo1.y = xo[o + 1280];
    v2f bn0; bn0.x = xn[o];        bn0.y = xn[o + 256];
    v2f bn1; bn1.x = xn[o + 1024]; bn1.y = xn[o + 1280];
    co = __builtin_amdgcn_wmma_f32_16x16x4_f32(false, a0, false, bo0, (short)0, co, false, false);
    cn = __builtin_amdgcn_wmma_f32_16x16x4_f32(false, a0, false, bn0, (short)0, cn, false, false);
    co = __builtin_amdgcn_wmma_f32_16x16x4_f32(false, a1, false, bo1, (short)0, co, false, false);
    cn = __builtin_amdgcn_wmma_f32_16x16x4_f32(false, a1, false, bn1, (short)0, cn, false, false);
  }

  // spill each wave's 16x16 tiles to LDS
  for (int r = 0; r < 8; ++r) {
    int m = r + (hi << 3);
    red[0][wave][m * 16 + j] = co[r];
    red[1][wave][m * 16 + j] = cn[r];
  }
  __syncthreads();

  // fixed-order reduce across the 8 waves; 256 tile elems == 256 threads
  int t = threadIdx.x;
  float so = 0.0f, sn = 0.0f;
  for (int w = 0; w < 8; ++w) { so += red[0][w][t]; sn += red[1][w][t]; }
  size_t off = (size_t)(tileBase + (int)blockIdx.x) * 256 + t;
  S_old[off] = so;
  S_new[off] = sn;
}

// ---------------------------------------------------------------
// Kernel 3: finalize. 288 vectors of 256 channels; one block,
// fixed-order LDS tree reductions (deterministic).  1/cnt cancels under
// l2-normalization so S tiles are used directly.
// ---------------------------------------------------------------
__global__ __launch_bounds__(256)
void sat_finalize(const float* __restrict__ S_old, const float* __restrict__ S_new,
                  float* __restrict__ out) {
  const int HS[4] = {2, 3, 5, 8};
  const int TB[4] = {0, 64, 160, 320};
  __shared__ float sh[256];
  __shared__ float bc[2];
  int t  = threadIdx.x;
  int kt = t >> 4;
  int j  = t & 15;
  float total = 0.0f;

  for (int s = 0; s < 4; ++s) {
    int H = HS[s];
    float ssum = 0.0f;
    for (int b = 0; b < 2; ++b)
      for (int h = 0; h < H; ++h)
        for (int n = 0; n < 8; ++n) {
          size_t idx = (size_t)(TB[s] + (b * H + h) * 16 + kt) * 256 + n * 16 + j;
          float vo = S_old[idx];
          float vn = S_new[idx];

          sh[t] = vo * vo; __syncthreads();
          for (int st = 128; st > 0; st >>= 1) { if (t < st) sh[t] += sh[t + st]; __syncthreads(); }
          if (t == 0) bc[0] = fmaxf(sqrtf(sh[0]), 1e-12f);
          __syncthreads();

          sh[t] = vn * vn; __syncthreads();
          for (int st = 128; st > 0; st >>= 1) { if (t < st) sh[t] += sh[t + st]; __syncthreads(); }
          if (t == 0) bc[1] = fmaxf(sqrtf(sh[0]), 1e-12f);
          __syncthreads();

          float d = vo / bc[0] - vn / bc[1];
          sh[t] = d * d; __syncthreads();
          for (int st = 128; st > 0; st >>= 1) { if (t < st) sh[t] += sh[t + st]; __syncthreads(); }
          if (t == 0) ssum += sqrtf(sh[0]);
          __syncthreads();
        }
    if (t == 0) total += ssum / (float)(2 * 8 * H);
  }
  if (t == 0) out[0] = total * 0.25f;
}

// ---------------------------------------------------------------
extern "C" void kernel_launch(void* const* d_in, const int* in_sizes, int n_in,
                              void* d_out, int out_size, void* d_ws, size_t ws_size,
                              hipStream_t stream) {
  (void)in_sizes; (void)n_in; (void)out_size; (void)ws_size;
  // inputs: old0..3, new0..3, instance_mask (int/bool), bbox (unused)
  const int* d_mask = (const int*)d_in[8];

  float* ws    = (float*)d_ws;
  float* maskF = ws;
  float* S_old = ws + S_OLD_OFF;
  float* S_new = ws + S_NEW_OFF;

  sat_mask_prep<<<128, 256, 0, stream>>>(d_mask, maskF);

  const int Hs[4]  = {2, 3, 5, 8};
  const int Ls[4]  = {16384, 4096, 1024, 256};
  const int MBs[4] = {MB0, MB1, MB2, MB3};
  const int TBs[4] = {0, 64, 160, 320};
  for (int s = 0; s < 4; ++s) {
    int blocks = 2 * Hs[s] * 16;   // B * H * ktiles
    sat_wmma<<<blocks, 256, 0, stream>>>((const float*)d_in[s], (const float*)d_in[4 + s],
                                         maskF + MBs[s], S_old, S_new,
                                         Hs[s], Ls[s], TBs[s]);
  }

  sat_finalize<<<1, 256, 0, stream>>>(S_old, S_new, (float*)d_out);
}